// TPAttention_3092376453855
// MI455X (gfx1250) — compile-verified
//
#include <hip/hip_runtime.h>
#include <hip/hip_bf16.h>
#include <math.h>

// ---------------------------------------------------------------------------
// Problem constants (from reference)
// ---------------------------------------------------------------------------
#define D_MODEL   1024
#define N_HEADS   16
#define D_KV      64
#define INNER     1024
#define NUM_ROLES 64
#define NUM_BUCK  32
#define MAX_DIST  128
#define BS        4
#define QLEN      1024
#define MROWS     (BS * QLEN)     // 4096

typedef __attribute__((ext_vector_type(2))) float v2f;
typedef __attribute__((ext_vector_type(8))) float v8f;
typedef __attribute__((ext_vector_type(4))) int   v4i;

// f32 WMMA: D(16x16) = A(16x4) * B(4x16) + C, wave32.
__device__ __forceinline__ v8f wmma_f32(v2f a, v2f b, v8f c) {
    return __builtin_amdgcn_wmma_f32_16x16x4_f32(
        /*neg_a=*/false, a, /*neg_b=*/false, b,
        /*c_mod=*/(short)0, c, /*reuse_a=*/false, /*reuse_b=*/false);
}

// ---------------------------------------------------------------------------
// Async global->LDS copy (CDNA5 GLOBAL_LOAD_ASYNC_TO_LDS_B128, ASYNCcnt),
// with a VGPR-roundtrip fallback if the builtins are not declared.
// Builtin signature (from compiler diagnostic): param0 = v4i in AS1,
// param1 = LDS dest, 2 int immediates (offset, cpol).
// ---------------------------------------------------------------------------
#if defined(__has_builtin)
#  if __has_builtin(__builtin_amdgcn_global_load_async_to_lds_b128) && \
      __has_builtin(__builtin_amdgcn_s_wait_asynccnt)
#    define HAVE_ASYNC_LDS 1
#  endif
#endif
#ifndef HAVE_ASYNC_LDS
#  define HAVE_ASYNC_LDS 0
#endif

typedef __attribute__((address_space(1))) v4i g_v4i;
typedef __attribute__((address_space(3))) v4i l_v4i;

__device__ __forceinline__ void cp16_to_lds(const float* g, float* l) {
#if HAVE_ASYNC_LDS
    __builtin_amdgcn_global_load_async_to_lds_b128(
        (g_v4i*)g, (l_v4i*)l, 0, 0);
#else
    *reinterpret_cast<float4*>(l) = *reinterpret_cast<const float4*>(g);
#endif
}

__device__ __forceinline__ void cp_wait() {
#if HAVE_ASYNC_LDS
    __builtin_amdgcn_s_wait_asynccnt(0);
#endif
}

// ---------------------------------------------------------------------------
// T5 bidirectional relative-position bucket (rel = mem_pos - ctx_pos)
// ---------------------------------------------------------------------------
__device__ __forceinline__ int rel_bucket(int rel) {
    int n = -rel;
    const int nb = NUM_BUCK / 2;        // 16
    int ret = (n < 0) ? nb : 0;
    n = abs(n);
    const int max_exact = nb / 2;       // 8
    if (n < max_exact) return ret + n;
    float v = logf((float)max(n, 1) / (float)max_exact) /
              logf((float)MAX_DIST / (float)max_exact) * (float)(nb - max_exact);
    int vi = max_exact + (int)v;
    vi = min(vi, nb - 1);
    return ret + vi;
}

// ---------------------------------------------------------------------------
// Normalize role matrix rows: Rn[r][d] = R[r][d] / ||R[r]||
// ---------------------------------------------------------------------------
__global__ void rn_kernel(const float* __restrict__ R, float* __restrict__ Rn) {
    int r = threadIdx.x;                // 64 threads
    float ss = 0.f;
    #pragma unroll 8
    for (int d = 0; d < D_KV; ++d) { float x = R[r * D_KV + d]; ss += x * x; }
    float inv = 1.0f / sqrtf(ss);
    #pragma unroll 8
    for (int d = 0; d < D_KV; ++d) Rn[r * D_KV + d] = R[r * D_KV + d] * inv;
}

// ---------------------------------------------------------------------------
// Generic fp32 WMMA GEMM: C[M,N] = A[M,K] * B[K,N]  (row-major)
// Block: 256 threads = 8 waves arranged 2(M) x 4(N); block tile 64x128.
// Each wave computes a 32x32 tile (2x2 accumulators) -> 2x fragment reuse,
// 4 independent WMMAs per LDS fetch group. K staged in 32-chunks through
// double-buffered LDS filled with async global->LDS copies.
// mode 0: plain row-major store
// mode 1: QKV shaped store -> out[((b*16+h)*QLEN+q)*64+d]
// mode 2: plain + bias[col]
// ---------------------------------------------------------------------------
#define KC   32
#define APAD 36   // 32 + 4: keeps 16B alignment, 16 distinct banks for A reads
#define BPAD 132  // 128 + 4

__global__ __launch_bounds__(256) void gemm_f32_kernel(
    const float* __restrict__ A, const float* __restrict__ B,
    const float* __restrict__ bias, float* __restrict__ C,
    int M, int N, int K, int mode)
{
    __shared__ float As[2][64][APAD];   // 18.0 KB
    __shared__ float Bs[2][KC][BPAD];   // 33.0 KB

    const int tid    = threadIdx.x;
    const int wid    = tid >> 5;
    const int lane   = tid & 31;
    const int lane15 = lane & 15;
    const int hi     = lane >> 4;
    const int waveM  = wid >> 2;        // 0..1  -> 32-row slice
    const int waveN  = wid & 3;         // 0..3  -> 32-col slice
    const int row0   = blockIdx.y * 64;
    const int col0   = blockIdx.x * 128;

    v8f acc[2][2];
    #pragma unroll
    for (int i = 0; i < 2; ++i)
        #pragma unroll
        for (int j = 0; j < 2; ++j)
            acc[i][j] = (v8f){0.f,0.f,0.f,0.f,0.f,0.f,0.f,0.f};

    // ---- stage one K-chunk into LDS buffer `buf` ----
    auto stage = [&](int buf, int kk) {
        // A tile 64x32 : 512 float4, 2 per thread
        #pragma unroll
        for (int j = 0; j < 2; ++j) {
            int f4 = tid * 2 + j;
            int r  = f4 >> 3;           // 0..63
            int c  = (f4 & 7) * 4;      // 0..28
            cp16_to_lds(A + (size_t)(row0 + r) * K + kk + c, &As[buf][r][c]);
        }
        // B tile 32x128 : 1024 float4, 4 per thread
        #pragma unroll
        for (int j = 0; j < 4; ++j) {
            int f4 = tid * 4 + j;
            int r  = f4 >> 5;           // 0..31
            int c  = (f4 & 31) * 4;     // 0..124
            cp16_to_lds(B + (size_t)(kk + r) * N + col0 + c, &Bs[buf][r][c]);
        }
    };

    int buf = 0;
    stage(0, 0);

    for (int kk = 0; kk < K; kk += KC) {
        cp_wait();
        __syncthreads();
        if (kk + KC < K) stage(buf ^ 1, kk + KC);   // overlap with compute

        const int am0 = waveM * 32 + lane15;
        const int bn0 = waveN * 32 + lane15;
        #pragma unroll
        for (int k2 = 0; k2 < KC; k2 += 4) {
            v2f a0, a1, b0, b1;
            a0.x = As[buf][am0     ][k2 + hi * 2];
            a0.y = As[buf][am0     ][k2 + hi * 2 + 1];
            a1.x = As[buf][am0 + 16][k2 + hi * 2];
            a1.y = As[buf][am0 + 16][k2 + hi * 2 + 1];
            b0.x = Bs[buf][k2 + hi * 2    ][bn0];
            b0.y = Bs[buf][k2 + hi * 2 + 1][bn0];
            b1.x = Bs[buf][k2 + hi * 2    ][bn0 + 16];
            b1.y = Bs[buf][k2 + hi * 2 + 1][bn0 + 16];
            acc[0][0] = wmma_f32(a0, b0, acc[0][0]);
            acc[0][1] = wmma_f32(a0, b1, acc[0][1]);
            acc[1][0] = wmma_f32(a1, b0, acc[1][0]);
            acc[1][1] = wmma_f32(a1, b1, acc[1][1]);
        }
        buf ^= 1;
    }

    // ---- epilogue ----
    #pragma unroll
    for (int sm = 0; sm < 2; ++sm) {
        #pragma unroll
        for (int sn = 0; sn < 2; ++sn) {
            const int colw = col0 + waveN * 32 + sn * 16 + lane15;
            #pragma unroll
            for (int v = 0; v < 8; ++v) {
                int row = row0 + waveM * 32 + sm * 16 + v + hi * 8;
                float val = acc[sm][sn][v];
                if (mode == 2) val += bias[colw];
                if (mode == 1) {
                    int b = row >> 10, q = row & (QLEN - 1);
                    int h = colw >> 6, d = colw & 63;
                    C[((size_t)((b * N_HEADS + h) * QLEN + q)) * D_KV + d] = val;
                } else {
                    C[(size_t)row * N + colw] = val;
                }
            }
        }
    }
}

// ---------------------------------------------------------------------------
// Fused attention: per (b,h), flash-style.
// Block = 128 threads = 4 waves; each wave owns 16 q-rows.
// Q/K/V stored per-head: [bh][q][d] (bh = b*16+h).
// Output v_bar written UNSHAPED: vbar_un[(b*QLEN+q)*INNER + h*64 + d].
// ---------------------------------------------------------------------------
__global__ __launch_bounds__(128) void attn_kernel(
    const float* __restrict__ Qm, const float* __restrict__ Km,
    const float* __restrict__ Vm, const float* __restrict__ rel_bias,
    float* __restrict__ vbar_un)
{
    __shared__ float biasS[NUM_BUCK * N_HEADS];   // 2 KB
    __shared__ float pbuf[4][16][17];             // P tile bounce, per wave

    const int tid = threadIdx.x;
    for (int i = tid; i < NUM_BUCK * N_HEADS; i += 128) biasS[i] = rel_bias[i];
    __syncthreads();

    const int wid    = tid >> 5;
    const int lane   = tid & 31;
    const int lane15 = lane & 15;
    const int hi     = lane >> 4;
    const int bh     = blockIdx.y;
    const int b      = bh >> 4;
    const int h      = bh & 15;
    const int q0     = blockIdx.x * 64 + wid * 16;

    const float* Qp = Qm + (size_t)bh * QLEN * D_KV;
    const float* Kp = Km + (size_t)bh * QLEN * D_KV;
    const float* Vp = Vm + (size_t)bh * QLEN * D_KV;

    // Resident Q fragments: A-layout for all 16 K-chunks of d (16x64).
    v2f qa[16];
    #pragma unroll
    for (int c = 0; c < 16; ++c) {
        const float* p = Qp + (size_t)(q0 + lane15) * D_KV + 4 * c + hi * 2;
        qa[c].x = p[0];
        qa[c].y = p[1];
    }

    float m[8], l[8];
    v8f o[4];
    #pragma unroll
    for (int v = 0; v < 8; ++v) { m[v] = -3.0e38f; l[v] = 0.f; }
    #pragma unroll
    for (int t = 0; t < 4; ++t) o[t] = (v8f){0.f,0.f,0.f,0.f,0.f,0.f,0.f,0.f};

    for (int k0 = 0; k0 < QLEN; k0 += 16) {
        // ---- S(16x16) = Q(16x64) . K^T(64x16) : 16 wmma over d ----
        v8f s = {0.f,0.f,0.f,0.f,0.f,0.f,0.f,0.f};
        #pragma unroll
        for (int c = 0; c < 16; ++c) {
            v2f bk;
            const float* p = Kp + (size_t)(k0 + lane15) * D_KV + 4 * c + hi * 2;
            bk.x = p[0];
            bk.y = p[1];
            s = wmma_f32(qa[c], bk, s);
        }

        // ---- relative position bias ----
        const int kcol = k0 + lane15;
        #pragma unroll
        for (int v = 0; v < 8; ++v) {
            int qrow = q0 + v + hi * 8;
            s[v] += biasS[rel_bucket(kcol - qrow) * N_HEADS + h];
        }

        // ---- online softmax (rows live within 16-lane halves) ----
        #pragma unroll
        for (int v = 0; v < 8; ++v) {
            float rmax = s[v];
            rmax = fmaxf(rmax, __shfl_xor(rmax, 1));
            rmax = fmaxf(rmax, __shfl_xor(rmax, 2));
            rmax = fmaxf(rmax, __shfl_xor(rmax, 4));
            rmax = fmaxf(rmax, __shfl_xor(rmax, 8));
            float mn   = fmaxf(m[v], rmax);
            float pexp = __expf(s[v] - mn);
            float rsum = pexp;
            rsum += __shfl_xor(rsum, 1);
            rsum += __shfl_xor(rsum, 2);
            rsum += __shfl_xor(rsum, 4);
            rsum += __shfl_xor(rsum, 8);
            float alpha = __expf(m[v] - mn);
            l[v] = l[v] * alpha + rsum;
            m[v] = mn;
            #pragma unroll
            for (int t = 0; t < 4; ++t) o[t][v] *= alpha;
            // stash P into LDS in C-layout for re-shaping into A fragments
            pbuf[wid][v + hi * 8][lane15] = pexp;
        }

        // ---- O(16x64) += P(16x16) . V(16x64) : 4 k-chunks x 4 d-tiles ----
        #pragma unroll
        for (int cc = 0; cc < 4; ++cc) {
            v2f pa;
            pa.x = pbuf[wid][lane15][4 * cc + hi * 2];
            pa.y = pbuf[wid][lane15][4 * cc + hi * 2 + 1];
            #pragma unroll
            for (int t = 0; t < 4; ++t) {
                v2f bv;
                const float* vp = Vp + (size_t)(k0 + 4 * cc + hi * 2) * D_KV
                                     + 16 * t + lane15;
                bv.x = vp[0];
                bv.y = vp[D_KV];
                o[t] = wmma_f32(pa, bv, o[t]);
            }
        }
    }

    // ---- epilogue: v_bar = O / l, written unshaped [s, h*64+d] ----
    #pragma unroll
    for (int v = 0; v < 8; ++v) {
        float inv = 1.0f / l[v];
        int qrow = q0 + v + hi * 8;
        size_t base = ((size_t)(b * QLEN + qrow)) * INNER + h * D_KV;
        #pragma unroll
        for (int t = 0; t < 4; ++t)
            vbar_un[base + 16 * t + lane15] = o[t][v] * inv;
    }
}

// ---------------------------------------------------------------------------
// Role softmax + role mixing + Hadamard binding.
// One thread per (s, h) pair: softmax over 64 roles, roles = attn . Rn,
// ctx[s, h*64+d] = vbar[s, h*64+d] * roles[d].
// ---------------------------------------------------------------------------
__global__ __launch_bounds__(256) void role_bind_kernel(
    const float* __restrict__ RS, const float* __restrict__ Rn,
    const float* __restrict__ vbar_un, float* __restrict__ ctx)
{
    __shared__ float RnS[NUM_ROLES][D_KV + 1];
    const int tid = threadIdx.x;
    for (int i = tid; i < NUM_ROLES * D_KV; i += 256)
        RnS[i >> 6][i & 63] = Rn[i];
    __syncthreads();

    const int gid = blockIdx.x * 256 + tid;       // 0..65535
    const int s   = gid >> 4;                     // (b,q) row
    const int h   = gid & 15;

    const float* rs = RS + (size_t)s * INNER + h * NUM_ROLES;
    float a[NUM_ROLES];
    float mx = -3.0e38f;
    #pragma unroll 8
    for (int r = 0; r < NUM_ROLES; ++r) { a[r] = rs[r]; mx = fmaxf(mx, a[r]); }
    float sum = 0.f;
    #pragma unroll 8
    for (int r = 0; r < NUM_ROLES; ++r) { a[r] = __expf(a[r] - mx); sum += a[r]; }
    const float inv = 1.0f / sum;

    const float* vb = vbar_un + (size_t)s * INNER + h * D_KV;
    float*       co = ctx     + (size_t)s * INNER + h * D_KV;
    for (int d = 0; d < D_KV; ++d) {
        float acc = 0.f;
        #pragma unroll 8
        for (int r = 0; r < NUM_ROLES; ++r) acc += a[r] * RnS[r][d];
        co[d] = vb[d] * (acc * inv);
    }
}

// ---------------------------------------------------------------------------
// Host launcher
// ---------------------------------------------------------------------------
extern "C" void kernel_launch(void* const* d_in, const int* in_sizes, int n_in,
                              void* d_out, int out_size, void* d_ws, size_t ws_size,
                              hipStream_t stream) {
    const float* X        = (const float*)d_in[0];
    const float* Wq       = (const float*)d_in[1];
    const float* Wk       = (const float*)d_in[2];
    const float* Wv       = (const float*)d_in[3];
    const float* Wo       = (const float*)d_in[4];
    const float* Wr       = (const float*)d_in[5];
    const float* br       = (const float*)d_in[6];
    const float* R        = (const float*)d_in[7];
    const float* rel_bias = (const float*)d_in[8];

    float* ws = (float*)d_ws;
    const size_t SZ = (size_t)BS * N_HEADS * QLEN * D_KV;   // 4M floats / 16 MB
    float* Q    = ws;             // per-head layout
    float* Kb   = ws + 1 * SZ;
    float* Vb   = ws + 2 * SZ;
    float* Vbar = ws + 3 * SZ;    // unshaped [4096,1024]
    float* Rn   = ws + 4 * SZ;    // 64x64
    float* RS   = Q;              // reuse (Q dead after attention)
    float* CTX  = Kb;             // reuse (K dead after attention)

    dim3 gGemm(INNER / 128, MROWS / 64);   // (8, 64)

    rn_kernel<<<1, 64, 0, stream>>>(R, Rn);

    gemm_f32_kernel<<<gGemm, 256, 0, stream>>>(X, Wq, nullptr, Q,  MROWS, INNER, D_MODEL, 1);
    gemm_f32_kernel<<<gGemm, 256, 0, stream>>>(X, Wk, nullptr, Kb, MROWS, INNER, D_MODEL, 1);
    gemm_f32_kernel<<<gGemm, 256, 0, stream>>>(X, Wv, nullptr, Vb, MROWS, INNER, D_MODEL, 1);

    attn_kernel<<<dim3(QLEN / 64, BS * N_HEADS), 128, 0, stream>>>(Q, Kb, Vb, rel_bias, Vbar);

    gemm_f32_kernel<<<gGemm, 256, 0, stream>>>(Vbar, Wr, br, RS, MROWS, INNER, D_MODEL, 2);

    role_bind_kernel<<<(MROWS * N_HEADS) / 256, 256, 0, stream>>>(RS, Rn, Vbar, CTX);

    gemm_f32_kernel<<<gGemm, 256, 0, stream>>>(CTX, Wo, nullptr, (float*)d_out, MROWS, D_MODEL, INNER, 0);
}